// MahalanobisDistance_10496900072134
// MI455X (gfx1250) — compile-verified
//
#include <hip/hip_runtime.h>
#include <hip/hip_bf16.h>

typedef float v2f __attribute__((ext_vector_type(2)));
typedef float v8f __attribute__((ext_vector_type(8)));

#define B_ 4096
#define D_ 128
#define C_ 256

// ---------------------------------------------------------------------------
// Kernel 0: Asym = alpha + alpha^T   (64 KB, coalesced rows for the GEMM)
// ---------------------------------------------------------------------------
__global__ __launch_bounds__(256) void mahal_asym_kernel(
    const float* __restrict__ alpha, float* __restrict__ Asym) {
  const int i = blockIdx.x * 256 + threadIdx.x;  // 0 .. 16383
  const int r = i >> 7;
  const int c = i & 127;
  Asym[i] = alpha[r * D_ + c] + alpha[c * D_ + r];
}

// ---------------------------------------------------------------------------
// Kernel 1: mAm[c] = means[:,c]^T * alpha * means[:,c]   (tiny: 256 floats)
// ---------------------------------------------------------------------------
__global__ __launch_bounds__(128) void mahal_mam_kernel(
    const float* __restrict__ alpha, const float* __restrict__ means,
    float* __restrict__ mAm) {
  __shared__ float red[128];
  const int c = blockIdx.x;
  const int d = threadIdx.x;
  const float* arow = alpha + d * D_;
  float w = 0.f;
  for (int e = 0; e < D_; ++e) w += arow[e] * means[e * C_ + c];
  red[d] = w * means[d * C_ + c];
  __syncthreads();
  for (int s = 64; s > 0; s >>= 1) {
    if (d < s) red[d] += red[d + s];
    __syncthreads();
  }
  if (d == 0) mAm[c] = red[0];
}

// ---------------------------------------------------------------------------
// Kernel 2: per 16-row tile of X:
//   Y = Xtile @ Asym                     (WMMA f32 16x16x4, 8 waves x 32)
//   t1[b] = 0.5 * dot(X[b,:], Y[b,:])    ( = x^T A x )
//   Z = Ytile @ means                    (WMMA f32 16x16x4, 8 waves x 64)
//   out[b] = min_c ( t1[b] - Z[b,c] + mAm[c] )
// ---------------------------------------------------------------------------
__global__ __launch_bounds__(256) void mahal_main_kernel(
    const float* __restrict__ X, const float* __restrict__ means,
    const float* __restrict__ Asym, const float* __restrict__ mAm,
    float* __restrict__ out) {
  constexpr int LDP = D_ + 4;  // padded LDS stride (132) -> conflict-free
  __shared__ float Xs[16 * LDP];
  __shared__ float Ys[16 * LDP];
  __shared__ float tmp[256];
  __shared__ float t1s[16];
  __shared__ float pm[8][16];

  const int tid = threadIdx.x;
  const int wave = tid >> 5;   // 8 waves (wave32)
  const int lane = tid & 31;
  const int hn = lane & 15;    // index within 16-lane half
  const int hi = lane >> 4;    // which half
  const int rowbase = blockIdx.x * 16;

  // Stage X tile (16 x 128) into LDS
  for (int i = tid; i < 16 * D_; i += 256) {
    int r = i >> 7;
    int cc = i & 127;
    Xs[r * LDP + cc] = X[(rowbase + r) * D_ + cc];
  }
  __syncthreads();

  // ---- Phase 1: Y = Xs @ Asym; wave owns cols [16w, 16w+16) ---------------
  {
    const int cb = wave * 16;
    const int cg = cb + hn;
    v8f acc = {};
#pragma unroll 4
    for (int k0 = 0; k0 < D_; k0 += 4) {
      const int ka = k0 + 2 * hi;
      v2f a, b;
      // A fragment (16x4): lanes 0-15 -> K = k0,k0+1 ; lanes 16-31 -> k0+2,k0+3
      a.x = Xs[hn * LDP + ka];
      a.y = Xs[hn * LDP + ka + 1];
      // B fragment (4x16): coalesced rows of precomputed alpha + alpha^T
      b.x = Asym[ka * D_ + cg];
      b.y = Asym[(ka + 1) * D_ + cg];
      acc = __builtin_amdgcn_wmma_f32_16x16x4_f32(false, a, false, b,
                                                  (short)0, acc, false, false);
    }
    // D layout: VGPR r holds row r + 8*hi, col = hn
#pragma unroll
    for (int r = 0; r < 8; ++r) {
      Ys[(r + 8 * hi) * LDP + cb + hn] = acc[r];
    }
  }
  __syncthreads();

  // ---- t1[b] = 0.5 * dot(Xs[b,:], Ys[b,:]) --------------------------------
  {
    const int bb = tid >> 4;
    const int seg = tid & 15;
    float s = 0.f;
#pragma unroll
    for (int j = 0; j < 8; ++j) {
      const int dc = seg * 8 + j;
      s += Xs[bb * LDP + dc] * Ys[bb * LDP + dc];
    }
    tmp[tid] = s;
  }
  __syncthreads();
  if (tid < 16) {
    float s = 0.f;
#pragma unroll
    for (int j = 0; j < 16; ++j) s += tmp[tid * 16 + j];
    t1s[tid] = 0.5f * s;
  }
  __syncthreads();

  // ---- Phase 2: Z = Ys @ means; fuse q and row-min ------------------------
  {
    float qm[8];
#pragma unroll
    for (int r = 0; r < 8; ++r) qm[r] = __builtin_inff();

    for (int t = 0; t < 2; ++t) {
      const int cb = (wave * 2 + t) * 16;
      v8f acc = {};
#pragma unroll 4
      for (int k0 = 0; k0 < D_; k0 += 4) {
        const int ka = k0 + 2 * hi;
        v2f a, b;
        a.x = Ys[hn * LDP + ka];
        a.y = Ys[hn * LDP + ka + 1];
        b.x = means[ka * C_ + cb + hn];
        b.y = means[(ka + 1) * C_ + cb + hn];
        acc = __builtin_amdgcn_wmma_f32_16x16x4_f32(
            false, a, false, b, (short)0, acc, false, false);
      }
      const float mac = mAm[cb + hn];
#pragma unroll
      for (int r = 0; r < 8; ++r) {
        const float q = t1s[r + 8 * hi] - acc[r] + mac;
        qm[r] = fminf(qm[r], q);
      }
    }
    // min across the 16 columns (lanes within each half; xor<=8 stays in half)
#pragma unroll
    for (int r = 0; r < 8; ++r) {
      float v = qm[r];
      v = fminf(v, __shfl_xor(v, 1, 32));
      v = fminf(v, __shfl_xor(v, 2, 32));
      v = fminf(v, __shfl_xor(v, 4, 32));
      v = fminf(v, __shfl_xor(v, 8, 32));
      if (hn == 0) pm[wave][r + 8 * hi] = v;
    }
  }
  __syncthreads();

  if (tid < 16) {
    float m = pm[0][tid];
#pragma unroll
    for (int w2 = 1; w2 < 8; ++w2) m = fminf(m, pm[w2][tid]);
    out[rowbase + tid] = m;
  }
}

// ---------------------------------------------------------------------------
extern "C" void kernel_launch(void* const* d_in, const int* in_sizes, int n_in,
                              void* d_out, int out_size, void* d_ws,
                              size_t ws_size, hipStream_t stream) {
  const float* X     = (const float*)d_in[0];  // [4096,128]
  const float* means = (const float*)d_in[1];  // [128,256]
  const float* alpha = (const float*)d_in[2];  // [128,128]
  float* out  = (float*)d_out;                 // [4096]
  float* mAm  = (float*)d_ws;                  // 256 floats
  float* Asym = (float*)d_ws + 256;            // 16384 floats (64 KB)

  mahal_asym_kernel<<<(D_ * D_) / 256, 256, 0, stream>>>(alpha, Asym);
  mahal_mam_kernel<<<C_, 128, 0, stream>>>(alpha, means, mAm);
  mahal_main_kernel<<<B_ / 16, 256, 0, stream>>>(X, means, Asym, mAm, out);
}